// Custom_Decoder_21345987461775
// MI455X (gfx1250) — compile-verified
//
#include <hip/hip_runtime.h>

// Problem constants from the reference.
constexpr int B_     = 8;
constexpr int C_     = 2;
constexpr int FRAMES = 3999;
constexpr int E      = 512;   // K
constexpr int W_     = 16;    // N (one WMMA tile wide)
constexpr int STEP   = 8;     // overlap-add hop
constexpr int OUT_T  = STEP * (FRAMES - 1) + W_;   // 32000
constexpr int BC     = B_ * C_;                    // 16
constexpr int TILES  = (FRAMES + 15) / 16;         // 250 row-tiles per (b,c)
constexpr int NWAVES = BC * TILES;                 // 4000 waves total

constexpr int BDIM            = 256;               // 8 wave32 waves per block
constexpr int WAVES_PER_BLOCK = BDIM / 32;
constexpr int LDS_STRIDE      = E + 4;             // pad 4 floats -> conflict-free b64 reads

typedef float v2f __attribute__((ext_vector_type(2)));
typedef float v8f __attribute__((ext_vector_type(8)));

__global__ void zero_out_kernel(float* __restrict__ out, int n) {
  int i = blockIdx.x * blockDim.x + threadIdx.x;
  if (i < n) out[i] = 0.0f;
}

__global__ __launch_bounds__(BDIM) void decoder_wmma_f32_kernel(
    const float* __restrict__ dec,   // [BC, FRAMES, E]
    const float* __restrict__ wgt,   // [W, E]
    float* __restrict__ out) {       // [BC, OUT_T], pre-zeroed
  __shared__ float lds_w[W_ * LDS_STRIDE];

  // Stage the 32 KB weight matrix into LDS once per block (row-major, padded).
  for (int idx = threadIdx.x; idx < W_ * E; idx += BDIM) {
    int r = idx >> 9;          // / E (E == 512)
    int c = idx & (E - 1);
    lds_w[r * LDS_STRIDE + c] = wgt[idx];
  }
  __syncthreads();

  const int lane  = threadIdx.x & 31;
  const int wave  = threadIdx.x >> 5;
  const int gwave = blockIdx.x * WAVES_PER_BLOCK + wave;  // grid sized exactly: < NWAVES

  const int bc         = gwave / TILES;
  const int tile       = gwave % TILES;
  const int frame_base = tile * 16;

  // A-matrix 16x4 fp32 layout: lanes 0-15 hold M=lane, K = {kk, kk+1};
  // lanes 16-31 hold M=lane-16, K = {kk+2, kk+3}.
  const int m    = lane & 15;
  const int koff = (lane >> 4) << 1;                      // 0 or 2
  const int frame_ld = min(frame_base + m, FRAMES - 1);   // clamp tail tile (loads only)
  const float* __restrict__ arow = dec + (size_t)(bc * FRAMES + frame_ld) * E;

  // B-matrix 4x16: B[k][n] = weight[n][k]; lane holds column n = lane&15,
  // K pairs mirroring the A layout.
  const int n = lane & 15;
  const float* brow = lds_w + n * LDS_STRIDE;

  v8f acc0 = {};
  v8f acc1 = {};

#pragma unroll 4
  for (int kk = 0; kk < E; kk += 8) {
    // K-slice 0: [kk, kk+4)
    v2f a0 = __builtin_nontemporal_load((const v2f*)(arow + kk + koff));
    v2f b0 = *(const v2f*)(brow + kk + koff);
    acc0 = __builtin_amdgcn_wmma_f32_16x16x4_f32(
        /*neg_a=*/false, a0, /*neg_b=*/false, b0,
        /*c_mod=*/(short)0, acc0, /*reuse_a=*/false, /*reuse_b=*/false);
    // K-slice 1: [kk+4, kk+8) into the second accumulator (hides D->C latency)
    v2f a1 = __builtin_nontemporal_load((const v2f*)(arow + kk + 4 + koff));
    v2f b1 = *(const v2f*)(brow + kk + 4 + koff);
    acc1 = __builtin_amdgcn_wmma_f32_16x16x4_f32(
        false, a1, false, b1, (short)0, acc1, false, false);
  }

  // C/D layout: lane n = column (tap), VGPR v = row v (lanes 0-15) or v+8 (16-31).
  // Overlap-add scatter: sample t = frame*STEP + n. Overlap factor 2 -> atomics.
  const int mbase = (lane >> 4) << 3;                     // 0 or 8
  float* __restrict__ orow = out + (size_t)bc * OUT_T;
#pragma unroll
  for (int v = 0; v < 8; ++v) {
    int frame = frame_base + mbase + v;
    if (frame < FRAMES) {
      float val = acc0[v] + acc1[v];
      unsafeAtomicAdd(orow + frame * STEP + n, val);      // global_atomic_add_f32
    }
  }
}

extern "C" void kernel_launch(void* const* d_in, const int* in_sizes, int n_in,
                              void* d_out, int out_size, void* d_ws, size_t ws_size,
                              hipStream_t stream) {
  const float* dec = (const float*)d_in[0];   // [8,2,3999,512] fp32
  const float* wgt = (const float*)d_in[1];   // [16,512] fp32
  float* out = (float*)d_out;                 // [8,2,32000] fp32

  // 1) zero-init the scatter target (harness poisons d_out).
  zero_out_kernel<<<(out_size + BDIM - 1) / BDIM, BDIM, 0, stream>>>(out, out_size);

  // 2) fused GEMM (fp32 WMMA) + overlap-and-add scatter.
  int grid = NWAVES / WAVES_PER_BLOCK;        // 500 blocks * 8 waves = 4000 tiles
  decoder_wmma_f32_kernel<<<grid, BDIM, 0, stream>>>(dec, wgt, out);
}